// Pool_FPS_5806795784775
// MI455X (gfx1250) — compile-verified
//
#include <hip/hip_runtime.h>
#include <stdint.h>

#define B_      8
#define N_      32768
#define C_      128
#define K_      1024
#define THREADS 1024
#define PPT     (N_ / THREADS)   // 32 points per thread

// Dynamic LDS layout (per workgroup):
//   [0      , 262144)  sxy plane: float2[N] interleaved {x,y}
//   [262144 , +512 )   wkeys[2][32]  (double-buffered wave partial keys)
//   [262656 , +256 )   wz[2][32]     (double-buffered wave partial z)
#define RED_OFF    262144
#define SMEM_BYTES (RED_OFF + 1024)

extern "C" __global__ __launch_bounds__(THREADS, 1)
void fps_kernel(const float* __restrict__ xyz,
                float* __restrict__ out_static,   // [B,3,K]
                int*   __restrict__ idxbuf)       // [B,K]
{
    extern __shared__ char smem[];
    const float2* sxy = (const float2*)smem;
    unsigned long long (*wkeys)[32] = (unsigned long long (*)[32])(smem + RED_OFF);
    float              (*wz)[32]    = (float (*)[32])(smem + RED_OFF + 512);

    const int tid  = threadIdx.x;
    const int lane = tid & 31;
    const int wid  = tid >> 5;
    const int b    = blockIdx.x;

    const float* xp = xyz + (size_t)b * (3 * N_);
    const float* yp = xp + N_;
    const float* zp = xp + 2 * N_;

    // ---- stage x,y planes into LDS (interleaved float2) via CDNA5 async-to-LDS ----
    const unsigned s_base = (unsigned)(uintptr_t)smem;  // low 32b of shared aperture == LDS offset
    #pragma unroll
    for (int t = 0; t < PPT; ++t) {
        unsigned i  = (unsigned)(tid + t * THREADS);
        unsigned gx = i * 4u;               // byte offset within a coordinate plane
        unsigned lx = s_base + i * 8u;      // LDS slot for x
        unsigned ly = lx + 4u;              // LDS slot for y
        asm volatile("global_load_async_to_lds_b32 %0, %1, %2"
                     :: "v"(lx), "v"(gx), "s"(xp) : "memory");
        asm volatile("global_load_async_to_lds_b32 %0, %1, %2"
                     :: "v"(ly), "v"(gx), "s"(yp) : "memory");
    }

    // own z coords + running min-distances live in VGPRs
    float zr[PPT], dist[PPT];
    #pragma unroll
    for (int j = 0; j < PPT; ++j) {
        zr[j]   = zp[tid + (j << 10)];
        dist[j] = __builtin_inff();
    }

    asm volatile("s_wait_asynccnt 0x0" ::: "memory");
    __syncthreads();

    int    cur = 0;
    float2 c0  = sxy[0];
    float  cx  = c0.x;
    float  cy  = c0.y;
    float  cz  = zp[0];

    for (int k = 0; k < K_; ++k) {
        if (tid == 0) {
            idxbuf[b * K_ + k] = cur;
            out_static[(size_t)b * (3 * K_) + k]          = cx;
            out_static[(size_t)b * (3 * K_) + K_ + k]     = cy;
            out_static[(size_t)b * (3 * K_) + 2 * K_ + k] = cz;
        }

        // update min-dists for my 32 points, track local argmax (first-occurrence ties)
        float bd = -1.0f; int bp = 0; float bz = 0.0f;
        #pragma unroll
        for (int j = 0; j < PPT; ++j) {
            int    p  = tid + (j << 10);
            float2 xy = sxy[p];                 // one ds_load_b64, bank-conflict free
            float  dx = xy.x - cx;
            float  dy = xy.y - cy;
            float  dz = zr[j] - cz;
            float  d  = dx * dx + dy * dy + dz * dz;
            float  dm = fminf(dist[j], d);
            dist[j] = dm;
            if (dm > bd) { bd = dm; bp = p; bz = zr[j]; }
        }

        // key: dist bits (monotonic for d>=0) high, ~index low
        // => max(key) == max dist; ties -> smallest index (matches jnp.argmax)
        unsigned long long key =
            ((unsigned long long)__float_as_uint(bd) << 32) | (unsigned)(~(unsigned)bp);
        float zb = bz;

        // wave32 butterfly reduction (carry winner's z alongside)
        #pragma unroll
        for (int off = 16; off > 0; off >>= 1) {
            unsigned long long ok = __shfl_xor(key, off, 32);
            float              oz = __shfl_xor(zb,  off, 32);
            if (ok > key) { key = ok; zb = oz; }
        }

        // double-buffered cross-wave partials: single barrier per iteration
        const int pp = k & 1;
        if (lane == 0) { wkeys[pp][wid] = key; wz[pp][wid] = zb; }
        __syncthreads();

        // every wave redundantly reduces the 32 wave-partials (bitwise identical result)
        unsigned long long k2 = wkeys[pp][lane];
        float              z2 = wz[pp][lane];
        #pragma unroll
        for (int off = 16; off > 0; off >>= 1) {
            unsigned long long ok = __shfl_xor(k2, off, 32);
            float              oz = __shfl_xor(z2, off, 32);
            if (ok > k2) { k2 = ok; z2 = oz; }
        }

        cur = (int)(~(unsigned)(k2 & 0xFFFFFFFFu));
        cz  = z2;
        float2 cxy = sxy[cur];
        cx = cxy.x;
        cy = cxy.y;
    }
}

extern "C" __global__ __launch_bounds__(K_)
void gather_feat_kernel(const float* __restrict__ feat,   // [B,C,N]
                        const int*   __restrict__ idxbuf, // [B,K]
                        float*       __restrict__ outf)   // [B,C,K]
{
    int bc = blockIdx.x;          // 0 .. B*C-1
    int b  = bc >> 7;             // C_ == 128
    int kk = threadIdx.x;         // 0 .. K-1
    int p  = idxbuf[b * K_ + kk];
    outf[(size_t)bc * K_ + kk] = feat[(size_t)bc * N_ + p];
}

extern "C" void kernel_launch(void* const* d_in, const int* in_sizes, int n_in,
                              void* d_out, int out_size, void* d_ws, size_t ws_size,
                              hipStream_t stream)
{
    const float* xyz  = (const float*)d_in[0];   // [B,3,N]
    const float* feat = (const float*)d_in[1];   // [B,C,N]

    float* out_static = (float*)d_out;                           // [B,3,K]
    float* out_feat   = (float*)d_out + (size_t)B_ * 3 * K_;     // [B,C,K]
    int*   idxbuf     = (int*)d_ws;                              // [B,K]

    fps_kernel<<<B_, THREADS, SMEM_BYTES, stream>>>(xyz, out_static, idxbuf);
    gather_feat_kernel<<<B_ * C_, K_, 0, stream>>>(feat, idxbuf, out_feat);
}